// BilateralFilter_22574348108604
// MI455X (gfx1250) — compile-verified
//
#include <hip/hip_runtime.h>
#include <stdint.h>

#define D_R   7
#define TILE  64
#define LW    (TILE + 2 * D_R)     // 78
#define IMG_H 768
#define IMG_W 768
#define NTAPJ (2 * D_R + 1)        // 15

// +0.5 / sigma_space^2 * log2(e) = (0.5/9)*1.4426950408889634
#define NC2F  (0.08014972449383130f)
// sqrt(0.5/sigma_color^2 * log2(e)) = sqrt(50*1.4426950408889634)
#define CSC   (8.4932178f)

__global__ __launch_bounds__(256) void bilateral_tile_kernel(
    const float* __restrict__ x, float* __restrict__ out)
{
    __shared__ float tile[LW * LW];            // 24336 B

    const int tid = threadIdx.x;
    const int tx0 = blockIdx.x * TILE;
    const int ty0 = blockIdx.y * TILE;
    const float* src = x   + (size_t)blockIdx.z * (IMG_H * IMG_W);
    float*       dst = out + (size_t)blockIdx.z * (IMG_H * IMG_W);

    // ---- Stage reflect-padded 78x78 tile into LDS via async global->LDS DMA
    // (ASYNCcnt-tracked; per-lane addresses carry the reflect border mapping).
    for (int e = tid; e < LW * LW; e += 256) {
        int ly = e / LW;
        int lx = e - ly * LW;
        int gy = ty0 - D_R + ly;
        int gx = tx0 - D_R + lx;
        gy = (gy < 0) ? -gy : ((gy >= IMG_H) ? (2 * IMG_H - 2 - gy) : gy);
        gx = (gx < 0) ? -gx : ((gx >= IMG_W) ? (2 * IMG_W - 2 - gx) : gx);
        uint64_t ga = (uint64_t)(uintptr_t)(src + gy * IMG_W + gx);
        uint32_t la = (uint32_t)(uintptr_t)(tile + e);
        asm volatile("global_load_async_to_lds_b32 %0, %1, off"
                     :: "v"(la), "v"(ga) : "memory");
    }
    asm volatile("s_wait_asynccnt 0x0" ::: "memory");
    __syncthreads();

    // Separable spatial-weight normalization: 1 / (row sum)^2.
    float rowsum = 0.f;
    #pragma unroll
    for (int i = -D_R; i <= D_R; ++i)
        rowsum += __builtin_amdgcn_exp2f(-NC2F * (float)(i * i));
    const float invS = 1.0f / (rowsum * rowsum);

    // 16 pixels per thread, strided by 256 -> conflict-free LDS, coalesced stores.
    int   cidx[16];
    float xc[16], mcx[16], acc[16];
    #pragma unroll
    for (int k = 0; k < 16; ++k) {
        int p  = tid + (k << 8);
        int py = p >> 6;
        int px = p & 63;
        cidx[k] = (py + D_R) * LW + (px + D_R);
        xc[k]   = tile[cidx[k]];
        mcx[k]  = -CSC * xc[k];
        acc[k]  = 0.f;
    }

    // 225 taps. Spatial weight folded into the exponent (no exp2 for it):
    //   w = exp2( -( (c*(s-x))^2 + NC2F*(i^2+j^2) ) ),  contribution = w*s.
    // 3 FMA + 1 v_exp per tap-pixel; j fully unrolled so tap loads use
    // immediate DS offsets (pairable into ds_load_2addr_b32).
    #pragma unroll 1
    for (int i = -D_R; i <= D_R; ++i) {
        const float bi = NC2F * (float)(i * i);
        float nls[NTAPJ];                       // k-invariant, lives in VGPRs
        #pragma unroll
        for (int j = 0; j < NTAPJ; ++j) {
            const int jj = j - D_R;
            nls[j] = bi + NC2F * (float)(jj * jj);
        }
        const int roff = i * LW - D_R;          // row base: offsets stay >= 0
        #pragma unroll
        for (int k = 0; k < 16; ++k) {
            const float* rowp = &tile[cidx[k] + roff];
            #pragma unroll
            for (int j = 0; j < NTAPJ; ++j) {
                float s  = rowp[j];
                float dp = __builtin_fmaf(CSC, s, mcx[k]);
                float t  = __builtin_fmaf(dp, dp, nls[j]);
                float w  = __builtin_amdgcn_exp2f(-t);   // neg -> src modifier
                acc[k]   = __builtin_fmaf(w, s, acc[k]);
            }
        }
    }

    // Remove center tap (its unnormalized contribution is exactly xc), normalize.
    #pragma unroll
    for (int k = 0; k < 16; ++k) {
        int p  = tid + (k << 8);
        int py = p >> 6;
        int px = p & 63;
        dst[(size_t)(ty0 + py) * IMG_W + (tx0 + px)] = invS * (acc[k] - xc[k]);
    }
}

extern "C" void kernel_launch(void* const* d_in, const int* in_sizes, int n_in,
                              void* d_out, int out_size, void* d_ws, size_t ws_size,
                              hipStream_t stream)
{
    const float* x   = (const float*)d_in[0];
    float*       out = (float*)d_out;
    const int planes = in_sizes[0] / (IMG_H * IMG_W);   // 16*3 = 48
    dim3 grid(IMG_W / TILE, IMG_H / TILE, planes);      // 12 x 12 x 48
    bilateral_tile_kernel<<<grid, dim3(256, 1, 1), 0, stream>>>(x, out);
}